// StyleCodebook_16587163697604
// MI455X (gfx1250) — compile-verified
//
#include <hip/hip_runtime.h>
#include <hip/hip_bf16.h>

// Problem constants (match reference)
#define BB 16
#define NN 4096
#define DD 256
#define KK 512
#define MM (BB * NN)          // 65536 rows
#define MT 128                // rows per workgroup (8 waves x 16)
#define LSTR 260              // padded LDS row stride (floats) -> conflict-free frags
#define CHUNK 64              // codebook columns per chunk

typedef __attribute__((ext_vector_type(2))) float v2f;
typedef __attribute__((ext_vector_type(8))) float v8f;

// ---------------------------------------------------------------------------
// CDNA5 async memory->LDS helpers (ASYNCcnt-tracked, no VGPR round trip).
// Inline asm is portable across ROCm 7.2 / amdgpu-toolchain per bridge doc.
// ---------------------------------------------------------------------------
__device__ __forceinline__ void async_ld_b128(void* lds_dst, const void* gsrc) {
    // Low 32 bits of a generic LDS pointer are the wave-relative LDS byte
    // offset (ISA 10.2 aperture mapping), which is what VDST wants.
    unsigned int ldsoff = (unsigned int)(size_t)lds_dst;
    asm volatile("global_load_async_to_lds_b128 %0, %1, off"
                 :: "v"(ldsoff), "v"(gsrc)
                 : "memory");
}

__device__ __forceinline__ void wait_async0() {
    asm volatile("s_wait_asynccnt 0" ::: "memory");
}

// ---------------------------------------------------------------------------
// Kernel 0: e_sq[k] = sum_d codebook[k][d]^2
// ---------------------------------------------------------------------------
__global__ __launch_bounds__(256) void vq_esq_kernel(const float* __restrict__ cb,
                                                     float* __restrict__ esq) {
    int k = blockIdx.x * blockDim.x + threadIdx.x;
    if (k < KK) {
        const float4* row = (const float4*)(cb + (size_t)k * DD);
        float s = 0.0f;
        #pragma unroll 8
        for (int i = 0; i < DD / 4; ++i) {
            float4 v = row[i];
            s = fmaf(v.x, v.x, s);
            s = fmaf(v.y, v.y, s);
            s = fmaf(v.z, v.z, s);
            s = fmaf(v.w, v.w, s);
        }
        esq[k] = s;
    }
}

// ---------------------------------------------------------------------------
// Kernel 1: fp32 WMMA GEMM (scores = e_sq - 2 * z.cb) + per-row argmin.
// Each workgroup: 128 rows x all 512 codebook entries.
// Each wave: 16 rows; 8 chunks of 64 cols (double-buffered async staging);
// 4 C-tiles per chunk.
// ---------------------------------------------------------------------------
__device__ __forceinline__ void upd_argmin(const v8f& acc, int col,
                                           const float* __restrict__ esq,
                                           float* bv, int* bi) {
    float e = esq[col];  // per-lane scalar (col depends on lane&15)
    #pragma unroll
    for (int r = 0; r < 8; ++r) {
        float v = fmaf(-2.0f, acc[r], e);
        int ci = col;
        #pragma unroll
        for (int m = 1; m < 16; m <<= 1) {
            float ov = __shfl_xor(v, m);
            int   oi = __shfl_xor(ci, m);
            if (ov < v || (ov == v && oi < ci)) { v = ov; ci = oi; }
        }
        if (v < bv[r] || (v == bv[r] && ci < bi[r])) { bv[r] = v; bi[r] = ci; }
    }
}

__device__ __forceinline__ void stage_chunk_async(float* dstbuf,
                                                  const float* __restrict__ cb,
                                                  int chunk, int tid) {
    // 64 rows x 256 cols, 16 x b128 per thread, direct to LDS.
    #pragma unroll
    for (int i = 0; i < (CHUNK * DD / 4) / 256; ++i) {   // 16 iters
        int lin = tid + i * 256;
        int row = lin >> 6;                              // 64 float4 per row
        int c4  = lin & 63;
        async_ld_b128(&dstbuf[row * LSTR + c4 * 4],
                      cb + (size_t)(chunk * CHUNK + row) * DD + c4 * 4);
    }
}

__global__ __launch_bounds__(256) void vq_argmin_kernel(const float* __restrict__ z,
                                                        const float* __restrict__ cb,
                                                        const float* __restrict__ esq,
                                                        int* __restrict__ out_idx) {
    extern __shared__ float lds[];
    float* lz  = lds;                          // 128 x 260 floats (z tile)
    float* lc0 = lds + MT * LSTR;              // 64 x 260 floats (cb chunk buf 0)
    float* lc1 = lc0 + CHUNK * LSTR;           // 64 x 260 floats (cb chunk buf 1)

    const int tid = threadIdx.x;
    const int m0  = blockIdx.x * MT;

    // Async-stage z tile: 128 rows x 256 cols (32 x b128 per thread).
    #pragma unroll
    for (int i = 0; i < (MT * DD / 4) / 256; ++i) {      // 32 iters
        int lin = tid + i * 256;
        int row = lin >> 6;
        int c4  = lin & 63;
        async_ld_b128(&lz[row * LSTR + c4 * 4],
                      z + (size_t)(m0 + row) * DD + c4 * 4);
    }
    // Async-stage codebook chunk 0 into buffer 0.
    stage_chunk_async(lc0, cb, 0, tid);
    wait_async0();
    __syncthreads();

    const int lane = tid & 31;
    const int wv   = tid >> 5;           // wave id 0..7
    const int mrow = lane & 15;
    const int koff = (lane < 16) ? 0 : 2;

    float bestv[8];
    int   besti[8];
    #pragma unroll
    for (int r = 0; r < 8; ++r) { bestv[r] = 3.402823466e38f; besti[r] = 0; }

    const float* za = &lz[(wv * 16 + mrow) * LSTR + koff];

    for (int chunk = 0; chunk < 8; ++chunk) {
        float* cur = (chunk & 1) ? lc1 : lc0;
        float* nxt = (chunk & 1) ? lc0 : lc1;

        // Prefetch next chunk into the other buffer; the barrier that ended
        // the previous iteration guarantees nobody still reads `nxt`.
        if (chunk < 7) stage_chunk_async(nxt, cb, chunk + 1, tid);

        v8f acc0 = {}, acc1 = {}, acc2 = {}, acc3 = {};
        const float* b0 = &cur[(0 * 16 + mrow) * LSTR + koff];
        const float* b1 = &cur[(1 * 16 + mrow) * LSTR + koff];
        const float* b2 = &cur[(2 * 16 + mrow) * LSTR + koff];
        const float* b3 = &cur[(3 * 16 + mrow) * LSTR + koff];

        #pragma unroll 8
        for (int d = 0; d < DD; d += 4) {
            v2f a  = *(const v2f*)(za + d);
            v2f q0 = *(const v2f*)(b0 + d);
            v2f q1 = *(const v2f*)(b1 + d);
            v2f q2 = *(const v2f*)(b2 + d);
            v2f q3 = *(const v2f*)(b3 + d);
            acc0 = __builtin_amdgcn_wmma_f32_16x16x4_f32(false, a, false, q0,
                                                         (short)0, acc0, false, false);
            acc1 = __builtin_amdgcn_wmma_f32_16x16x4_f32(false, a, false, q1,
                                                         (short)0, acc1, false, false);
            acc2 = __builtin_amdgcn_wmma_f32_16x16x4_f32(false, a, false, q2,
                                                         (short)0, acc2, false, false);
            acc3 = __builtin_amdgcn_wmma_f32_16x16x4_f32(false, a, false, q3,
                                                         (short)0, acc3, false, false);
        }

        int cbase = chunk * CHUNK + mrow;
        upd_argmin(acc0, cbase + 0 * 16, esq, bestv, besti);
        upd_argmin(acc1, cbase + 1 * 16, esq, bestv, besti);
        upd_argmin(acc2, cbase + 2 * 16, esq, bestv, besti);
        upd_argmin(acc3, cbase + 3 * 16, esq, bestv, besti);

        if (chunk < 7) {
            wait_async0();     // own async portion of chunk+1 landed in LDS
            __syncthreads();   // everyone's portion landed; `cur` free to reuse
        }
    }

    // C-tile layout: lanes 0-15 VGPR r -> M=r; lanes 16-31 VGPR r -> M=r+8.
    if (mrow == 0) {
        int rb = (lane < 16) ? 0 : 8;
        #pragma unroll
        for (int r = 0; r < 8; ++r)
            out_idx[m0 + wv * 16 + rb + r] = besti[r];
    }
}

// ---------------------------------------------------------------------------
// Kernel 2: gather embed, masked quantized + index outputs, per-row loss partial
// ---------------------------------------------------------------------------
__global__ __launch_bounds__(256) void vq_gather_kernel(const float* __restrict__ z,
                                                        const float* __restrict__ cb,
                                                        const int* __restrict__ mask,
                                                        const int* __restrict__ idx,
                                                        float* __restrict__ out,
                                                        float* __restrict__ partials) {
    const int m = blockIdx.x;
    const int d = threadIdx.x;
    const int k = idx[m];
    float e  = cb[(size_t)k * DD + d];
    float zv = z[(size_t)m * DD + d];
    float df = e - zv;

    __shared__ float red[256];
    red[d] = df * df;
    __syncthreads();
    #pragma unroll
    for (int s = 128; s > 0; s >>= 1) {
        if (d < s) red[d] += red[d + s];
        __syncthreads();
    }

    bool mk = mask[m] != 0;
    out[(size_t)m * DD + d] = mk ? e : 0.0f;
    if (d == 0) {
        partials[m] = red[0];
        out[(size_t)MM * DD + m] = mk ? (float)k : -1.0f;  // indices region
    }
}

// ---------------------------------------------------------------------------
// Kernel 3: deterministic final reduction -> commit loss scalar
// ---------------------------------------------------------------------------
__global__ __launch_bounds__(256) void vq_loss_kernel(const float* __restrict__ partials,
                                                      float* __restrict__ out) {
    __shared__ float red[256];
    float s = 0.0f;
    for (int i = threadIdx.x; i < MM; i += 256) s += partials[i];
    red[threadIdx.x] = s;
    __syncthreads();
    #pragma unroll
    for (int st = 128; st > 0; st >>= 1) {
        if (threadIdx.x < st) red[threadIdx.x] += red[threadIdx.x + st];
        __syncthreads();
    }
    if (threadIdx.x == 0)
        out[(size_t)MM * DD + MM] = 0.25f * red[0] / (float)((size_t)MM * DD);
}

// ---------------------------------------------------------------------------
extern "C" void kernel_launch(void* const* d_in, const int* in_sizes, int n_in,
                              void* d_out, int out_size, void* d_ws, size_t ws_size,
                              hipStream_t stream) {
    const float* z    = (const float*)d_in[0];
    const int*   mask = (const int*)d_in[1];
    const float* cb   = (const float*)d_in[2];
    float*       out  = (float*)d_out;

    float* esq      = (float*)d_ws;                                    // 512 floats
    int*   idxbuf   = (int*)((char*)d_ws + KK * sizeof(float));        // 65536 ints
    float* partials = (float*)((char*)d_ws + KK * sizeof(float)
                                          + MM * sizeof(int));         // 65536 floats

    vq_esq_kernel<<<(KK + 255) / 256, 256, 0, stream>>>(cb, esq);

    size_t shmem = (size_t)(MT * LSTR + 2 * CHUNK * LSTR) * sizeof(float);  // ~260 KB
    vq_argmin_kernel<<<MM / MT, 256, shmem, stream>>>(z, cb, esq, idxbuf);

    vq_gather_kernel<<<MM, 256, 0, stream>>>(z, cb, mask, idxbuf, out, partials);

    vq_loss_kernel<<<1, 256, 0, stream>>>(partials, out);
}